// DCNv3_PyTorch_85160611545643
// MI455X (gfx1250) — compile-verified
//
#include <hip/hip_runtime.h>
#include <math.h>

// ---------------------------------------------------------------------------
// DCNv3 block for MI455X (gfx1250, wave32). All four GEMMs run on
// V_WMMA_F32_16X16X4_F32 (fp32 matrix pipe, exact precision); depthwise+LN+GELU
// and the bilinear-gather core are VALU kernels with 32-lane coalesced channel
// access. ~4.3 GFLOP / ~67MB traffic (~3us memory floor at 23.3 TB/s).
// ---------------------------------------------------------------------------

#define BB   8
#define CC   128
#define H0   62
#define W0   62
#define HH   64      // after pad=1 on the 1x1 conv
#define WW   64
#define HP   66      // x_proj padded for sampling
#define WP   66
#define GG   4
#define GC   32
#define PP   9

typedef __attribute__((ext_vector_type(2))) float v2f;
typedef __attribute__((ext_vector_type(8))) float v8f;

#define WMMA_F32(a, b, c) \
  __builtin_amdgcn_wmma_f32_16x16x4_f32(false, (a), false, (b), (short)0, (c), false, false)

static __device__ __forceinline__ v2f ld2(const float* p) {
  const float2 t = *(const float2*)p;
  v2f r; r.x = t.x; r.y = t.y; return r;
}

// ---------------------------------------------------------------------------
// K0: NCHW -> NHWC transpose of x   (so GEMM A-rows are contiguous)
// ---------------------------------------------------------------------------
__global__ void k_transpose_x(const float* __restrict__ x, float* __restrict__ xt) {
  int idx = blockIdx.x * 256 + threadIdx.x;
  if (idx >= BB * CC * H0 * W0) return;
  int w = idx % W0;
  int h = (idx / W0) % H0;
  int c = (idx / (W0 * H0)) % CC;
  int b = idx / (W0 * H0 * CC);
  xt[(((b * H0 + h) * W0 + w) << 7) + c] = x[idx];
}

// ---------------------------------------------------------------------------
// K1: y[p][co] = sum_k x[p][k] * conv_w[co][k] + conv_b[co]   (1x1 conv, pad=1)
// A: xt rows; border pixels read a guaranteed-zero page (branchless pointer
// select, keeps the inner loop free of exec-mask save/restore).
// B[k][n] = conv_w[n*128+k] (transposed). grid: 512 blocks x 128 thr (4 waves);
// wave = one 16-pixel M tile, N = 128 (8 WMMA tiles, 32 K-steps).
// ---------------------------------------------------------------------------
__global__ __launch_bounds__(128)
void k_conv1x1_wmma(const float* __restrict__ xt, const float* __restrict__ cw,
                    const float* __restrict__ cb, const float* __restrict__ zeropage,
                    float* __restrict__ y) {
  const int lane  = threadIdx.x & 31;
  const int mtile = blockIdx.x * 4 + (threadIdx.x >> 5);
  const int n     = lane & 15;       // N within tile / M for A-load
  const int kh    = lane >> 4;       // K-half selector

  const int pA = mtile * 16 + n;     // pixel of the A row this lane loads
  const int bA = pA >> 12, hA = (pA >> 6) & 63, wA = pA & 63;
  const bool valid = (hA >= 1) & (hA <= 62) & (wA >= 1) & (wA <= 62);
  const float* arow = valid ? (xt + (((bA * H0 + (hA - 1)) * W0 + (wA - 1)) << 7))
                            : zeropage;   // 128 zero floats (xpad[0..127])

  v8f acc[8];
#pragma unroll
  for (int nt = 0; nt < 8; ++nt)
#pragma unroll
    for (int r = 0; r < 8; ++r) acc[nt][r] = 0.f;

  for (int kk = 0; kk < 128; kk += 4) {
    v2f a = ld2(arow + kk + 2 * kh);
#pragma unroll
    for (int nt = 0; nt < 8; ++nt) {
      v2f bv = ld2(cw + ((nt * 16 + n) << 7) + kk + 2 * kh);
      acc[nt] = WMMA_F32(a, bv, acc[nt]);
    }
  }
#pragma unroll
  for (int nt = 0; nt < 8; ++nt) {
    const int col  = nt * 16 + n;
    const float bc = cb[col];
#pragma unroll
    for (int r = 0; r < 8; ++r) {
      const int pp = mtile * 16 + r + 8 * kh;   // D layout: M = r + 8*(lane>>4)
      y[(pp << 7) + col] = acc[nt][r] + bc;
    }
  }
}

// ---------------------------------------------------------------------------
// K2: x_proj = y @ in_proj_w + in_proj_b, stored into 66x66 zero-padded buffer
// B[k][n] = ipw[k*128+n] (row-major)
// ---------------------------------------------------------------------------
__global__ __launch_bounds__(128)
void k_inproj_wmma(const float* __restrict__ y, const float* __restrict__ ipw,
                   const float* __restrict__ ipb, float* __restrict__ xpad) {
  const int lane  = threadIdx.x & 31;
  const int mtile = blockIdx.x * 4 + (threadIdx.x >> 5);
  const int n  = lane & 15;
  const int kh = lane >> 4;
  const float* arow = y + ((mtile * 16 + n) << 7);

  v8f acc[8];
#pragma unroll
  for (int nt = 0; nt < 8; ++nt)
#pragma unroll
    for (int r = 0; r < 8; ++r) acc[nt][r] = 0.f;

  for (int kk = 0; kk < 128; kk += 4) {
    v2f a = ld2(arow + kk + 2 * kh);
    const int k0 = kk + 2 * kh;
#pragma unroll
    for (int nt = 0; nt < 8; ++nt) {
      const int col = nt * 16 + n;
      v2f bv; bv.x = ipw[(k0 << 7) + col]; bv.y = ipw[((k0 + 1) << 7) + col];
      acc[nt] = WMMA_F32(a, bv, acc[nt]);
    }
  }
#pragma unroll
  for (int nt = 0; nt < 8; ++nt) {
    const int col  = nt * 16 + n;
    const float bc = ipb[col];
#pragma unroll
    for (int r = 0; r < 8; ++r) {
      const int pp = mtile * 16 + r + 8 * kh;
      const int b = pp >> 12, h = (pp >> 6) & 63, w = pp & 63;
      xpad[(((b * HP + h + 1) * WP + w + 1) << 7) + col] = acc[nt][r] + bc;
    }
  }
}

// ---------------------------------------------------------------------------
// K3: depthwise 3x3 (pad 1) + LayerNorm(C) + exact GELU -> d   (NHWC)
// one block per pixel, 128 threads = channels
// ---------------------------------------------------------------------------
__global__ __launch_bounds__(128)
void k_dw_ln_gelu(const float* __restrict__ y, const float* __restrict__ dww,
                  const float* __restrict__ dwb, const float* __restrict__ lng,
                  const float* __restrict__ lnb, float* __restrict__ d) {
  const int c = threadIdx.x;
  const int pix = blockIdx.x;
  const int b = pix >> 12, h = (pix >> 6) & 63, w = pix & 63;

  float v = dwb[c];
#pragma unroll
  for (int kh = 0; kh < 3; ++kh) {
    const int hh = h + kh - 1;
    if (hh < 0 || hh >= HH) continue;
#pragma unroll
    for (int kw = 0; kw < 3; ++kw) {
      const int ww = w + kw - 1;
      if (ww < 0 || ww >= WW) continue;
      v += y[(((b << 6) + hh) * WW + ww) * CC + c] * dww[c * 9 + kh * 3 + kw];
    }
  }

  __shared__ float red[128];
  red[c] = v;
  __syncthreads();
  for (int s = 64; s > 0; s >>= 1) { if (c < s) red[c] += red[c + s]; __syncthreads(); }
  const float mu = red[0] * (1.f / 128.f);
  __syncthreads();
  const float dv = v - mu;
  red[c] = dv * dv;
  __syncthreads();
  for (int s = 64; s > 0; s >>= 1) { if (c < s) red[c] += red[c + s]; __syncthreads(); }
  const float var = red[0] * (1.f / 128.f);

  float t = dv * rsqrtf(var + 1e-5f) * lng[c] + lnb[c];
  t = 0.5f * t * (1.f + erff(t * 0.70710678118654752f));   // exact GELU
  d[pix * CC + c] = t;
}

// ---------------------------------------------------------------------------
// K4: offset = d @ off_w + off_b (N=72) ; mask logits = d @ mask_w + mask_b (N=36)
// skinny GEMMs; column guards are clamped-index + 0/1 multiply so all loads are
// in-bounds and branchless (no exec manipulation in the K loop).
// ---------------------------------------------------------------------------
__global__ __launch_bounds__(128)
void k_offmask_wmma(const float* __restrict__ d,
                    const float* __restrict__ ow, const float* __restrict__ ob,
                    const float* __restrict__ mw, const float* __restrict__ mb,
                    float* __restrict__ off, float* __restrict__ msk) {
  const int lane  = threadIdx.x & 31;
  const int mtile = blockIdx.x * 4 + (threadIdx.x >> 5);
  const int n  = lane & 15;
  const int kh = lane >> 4;
  const float* arow = d + ((mtile * 16 + n) << 7);

  v8f aco[5], acm[3];
#pragma unroll
  for (int nt = 0; nt < 5; ++nt)
#pragma unroll
    for (int r = 0; r < 8; ++r) aco[nt][r] = 0.f;
#pragma unroll
  for (int nt = 0; nt < 3; ++nt)
#pragma unroll
    for (int r = 0; r < 8; ++r) acm[nt][r] = 0.f;

  for (int kk = 0; kk < 128; kk += 4) {
    v2f a = ld2(arow + kk + 2 * kh);
    const int k0 = kk + 2 * kh;
#pragma unroll
    for (int nt = 0; nt < 5; ++nt) {                 // offset: N = 72
      const int col  = nt * 16 + n;
      const int colc = (col < 72) ? col : 71;        // always in-bounds
      const float sel = (col < 72) ? 1.f : 0.f;
      v2f bv;
      bv.x = ow[k0 * 72 + colc] * sel;
      bv.y = ow[(k0 + 1) * 72 + colc] * sel;
      aco[nt] = WMMA_F32(a, bv, aco[nt]);
    }
#pragma unroll
    for (int nt = 0; nt < 3; ++nt) {                 // mask: N = 36
      const int col  = nt * 16 + n;
      const int colc = (col < 36) ? col : 35;
      const float sel = (col < 36) ? 1.f : 0.f;
      v2f bv;
      bv.x = mw[k0 * 36 + colc] * sel;
      bv.y = mw[(k0 + 1) * 36 + colc] * sel;
      acm[nt] = WMMA_F32(a, bv, acm[nt]);
    }
  }
#pragma unroll
  for (int nt = 0; nt < 5; ++nt) {
    const int col = nt * 16 + n;
    if (col < 72) {
      const float bc = ob[col];
#pragma unroll
      for (int r = 0; r < 8; ++r)
        off[(mtile * 16 + r + 8 * kh) * 72 + col] = aco[nt][r] + bc;
    }
  }
#pragma unroll
  for (int nt = 0; nt < 3; ++nt) {
    const int col = nt * 16 + n;
    if (col < 36) {
      const float bc = mb[col];
#pragma unroll
      for (int r = 0; r < 8; ++r)
        msk[(mtile * 16 + r + 8 * kh) * 36 + col] = acm[nt][r] + bc;
    }
  }
}

// ---------------------------------------------------------------------------
// K5: DCNv3 core. One wave per (pixel q, group g); lane = channel in group.
// ix = w+1+klin[i]+dx, iy = h+1+klin[j]+dy against the zero-padded 66x66 grid.
// Softmax over the 9 points folded in. Gather-bound; 32-lane coalesced loads.
// ---------------------------------------------------------------------------
static __device__ __forceinline__ float fetchs(const float* img, float xi, float yi) {
  const bool ok = (xi >= 0.f) & (xi <= (float)(WP - 1)) & (yi >= 0.f) & (yi <= (float)(HP - 1));
  const int xc = (int)fminf(fmaxf(xi, 0.f), (float)(WP - 1));
  const int yc = (int)fminf(fmaxf(yi, 0.f), (float)(HP - 1));
  const float v = img[(yc * WP + xc) << 7];
  return ok ? v : 0.f;
}

__global__ __launch_bounds__(256)
void k_dcnv3_core(const float* __restrict__ xpad, const float* __restrict__ off,
                  const float* __restrict__ msk, float* __restrict__ dcn) {
  const int lane = threadIdx.x & 31;
  const int task = blockIdx.x * 8 + (threadIdx.x >> 5);   // q*G + g
  const int g = task & 3;
  const int q = task >> 2;
  const int b = q >> 12, h = (q >> 6) & 63, w = q & 63;

  const float* offp = off + q * 72 + g * 18;
  const float* mskp = msk + q * 36 + g * 9;
  const float* img  = xpad + ((b * HP * WP) << 7) + g * GC + lane;

  float e[PP];
  float mx = -1e30f;
#pragma unroll
  for (int p = 0; p < PP; ++p) { e[p] = mskp[p]; mx = fmaxf(mx, e[p]); }
  float s = 0.f;
#pragma unroll
  for (int p = 0; p < PP; ++p) { e[p] = expf(e[p] - mx); s += e[p]; }
  const float inv = 1.f / s;

  float acc = 0.f;
#pragma unroll
  for (int p = 0; p < PP; ++p) {
    const int i = p / 3, j = p % 3;                // meshgrid 'ij': x uses i, y uses j
    const float ix = (float)(w + i) + offp[2 * p];       // w + 1 + (i-1) + dx
    const float iy = (float)(h + j) + offp[2 * p + 1];   // h + 1 + (j-1) + dy
    const float x0 = floorf(ix), y0 = floorf(iy);
    const float wx1 = ix - x0, wx0 = 1.f - wx1;
    const float wy1 = iy - y0, wy0 = 1.f - wy1;
    float v = fetchs(img, x0,       y0)       * (wx0 * wy0)
            + fetchs(img, x0 + 1.f, y0)       * (wx1 * wy0)
            + fetchs(img, x0,       y0 + 1.f) * (wx0 * wy1)
            + fetchs(img, x0 + 1.f, y0 + 1.f) * (wx1 * wy1);
    acc += v * (e[p] * inv);
  }
  dcn[(q << 7) + g * GC + lane] = acc;
}

// ---------------------------------------------------------------------------
// K6: out = dcn @ out_proj_w + out_proj_b, stored transposed to NCHW d_out
// ---------------------------------------------------------------------------
__global__ __launch_bounds__(128)
void k_outproj_wmma(const float* __restrict__ dcn, const float* __restrict__ opw,
                    const float* __restrict__ opb, float* __restrict__ out) {
  const int lane  = threadIdx.x & 31;
  const int mtile = blockIdx.x * 4 + (threadIdx.x >> 5);
  const int n  = lane & 15;
  const int kh = lane >> 4;
  const float* arow = dcn + ((mtile * 16 + n) << 7);

  v8f acc[8];
#pragma unroll
  for (int nt = 0; nt < 8; ++nt)
#pragma unroll
    for (int r = 0; r < 8; ++r) acc[nt][r] = 0.f;

  for (int kk = 0; kk < 128; kk += 4) {
    v2f a = ld2(arow + kk + 2 * kh);
    const int k0 = kk + 2 * kh;
#pragma unroll
    for (int nt = 0; nt < 8; ++nt) {
      const int col = nt * 16 + n;
      v2f bv; bv.x = opw[(k0 << 7) + col]; bv.y = opw[((k0 + 1) << 7) + col];
      acc[nt] = WMMA_F32(a, bv, acc[nt]);
    }
  }
#pragma unroll
  for (int nt = 0; nt < 8; ++nt) {
    const int co  = nt * 16 + n;
    const float bc = opb[co];
#pragma unroll
    for (int r = 0; r < 8; ++r) {
      const int pp = mtile * 16 + r + 8 * kh;
      const int b = pp >> 12, hw = pp & 4095;
      out[((b << 7) + co) * 4096 + hw] = acc[nt][r] + bc;   // NCHW
    }
  }
}

// ---------------------------------------------------------------------------
// launch
// ---------------------------------------------------------------------------
extern "C" void kernel_launch(void* const* d_in, const int* in_sizes, int n_in,
                              void* d_out, int out_size, void* d_ws, size_t ws_size,
                              hipStream_t stream) {
  const float* x    = (const float*)d_in[0];
  const float* cw   = (const float*)d_in[1];
  const float* cb   = (const float*)d_in[2];
  const float* ipw  = (const float*)d_in[3];
  const float* ipb  = (const float*)d_in[4];
  const float* dww  = (const float*)d_in[5];
  const float* dwb  = (const float*)d_in[6];
  const float* lng  = (const float*)d_in[7];
  const float* lnb  = (const float*)d_in[8];
  const float* ow   = (const float*)d_in[9];
  const float* ob   = (const float*)d_in[10];
  const float* mw   = (const float*)d_in[11];
  const float* mb   = (const float*)d_in[12];
  const float* opw  = (const float*)d_in[13];
  const float* opb  = (const float*)d_in[14];
  float* out = (float*)d_out;

  // workspace layout (floats); overlays keep total at ~67 MB
  const size_t SZ_XT   = (size_t)BB * H0 * W0 * CC;  // 3,936,256
  const size_t SZ_Y    = (size_t)BB * HH * WW * CC;  // 4,194,304
  const size_t SZ_XPAD = (size_t)BB * HP * WP * CC;  // 4,460,544
  float* ws   = (float*)d_ws;
  float* xt   = ws;                       // dead after K1 -> reused by off/msk
  float* yb   = ws + SZ_XT;               // dead after K3 -> reused by dcn
  float* xpad = yb + SZ_Y;
  float* db   = xpad + SZ_XPAD;
  float* offb = xt;                       // 2,359,296 floats
  float* mskb = xt + (size_t)BB * HH * WW * 72;  // 1,179,648 floats (fits in xt)
  float* dcnb = yb;

  hipMemsetAsync(xpad, 0, SZ_XPAD * sizeof(float), stream);  // pad ring must be 0
  // xpad[0..127] stays zero forever (inproj writes start at offset 67*128):
  // it doubles as the zero page for border A-rows in the conv GEMM.

  const int NPIX = BB * HH * WW;           // 32768
  const int GEMM_BLOCKS = NPIX / 64;       // 512 (4 waves x 16-pixel tiles)

  k_transpose_x<<<(BB * CC * H0 * W0 + 255) / 256, 256, 0, stream>>>(x, xt);
  k_conv1x1_wmma<<<GEMM_BLOCKS, 128, 0, stream>>>(xt, cw, cb, xpad, yb);
  k_inproj_wmma<<<GEMM_BLOCKS, 128, 0, stream>>>(yb, ipw, ipb, xpad);
  k_dw_ln_gelu<<<NPIX, 128, 0, stream>>>(yb, dww, dwb, lng, lnb, db);
  k_offmask_wmma<<<GEMM_BLOCKS, 128, 0, stream>>>(db, ow, ob, mw, mb, offb, mskb);
  k_dcnv3_core<<<NPIX * GG / 8, 256, 0, stream>>>(xpad, offb, mskb, dcnb);
  k_outproj_wmma<<<GEMM_BLOCKS, 128, 0, stream>>>(dcnb, opw, opb, out);
}